// SenderAgent_25314537243115
// MI455X (gfx1250) — compile-verified
//
#include <hip/hip_runtime.h>
#include <hip/hip_bf16.h>

typedef float v2f __attribute__((ext_vector_type(2)));
typedef float v8f __attribute__((ext_vector_type(8)));

#define G_GRAPHS 64

__device__ __forceinline__ v8f wmma_k4(v2f a, v2f b, v8f c) {
  // D = A(16x4,f32) * B(4x16,f32) + C(16x16,f32) ; exact f32 matrix op
  return __builtin_amdgcn_wmma_f32_16x16x4_f32(false, a, false, b, (short)0, c,
                                               false, false);
}

__device__ __forceinline__ void atomicAddF32(float* p, float v) {
  __hip_atomic_fetch_add(p, v, __ATOMIC_RELAXED, __HIP_MEMORY_SCOPE_AGENT);
}

// ---------------------------------------------------------------- zero fill
__global__ __launch_bounds__(256) void zero_kernel(float* __restrict__ p, int n) {
  int i = blockIdx.x * 256 + threadIdx.x;
  if (i < n) p[i] = 0.0f;
}

// ------------------------------------------------------------- degree count
__global__ __launch_bounds__(256) void degree_kernel(const int* __restrict__ dst,
                                                     float* __restrict__ cnt, int nE) {
  int e = blockIdx.x * 256 + threadIdx.x;
  if (e < nE) atomicAddF32(&cnt[dst[e]], 1.0f);
}

// --------------------------------------------------- Y[N,64] = X[N,64] @ W[64,64]
// one wave per 16-row tile, W staged in LDS, f32 WMMA 16x16x4
__global__ __launch_bounds__(256) void gemm64_kernel(const float* __restrict__ X,
                                                     const float* __restrict__ W,
                                                     float* __restrict__ Y, int nRows) {
  __shared__ float sW[64 * 64];
  int t = threadIdx.x;
  for (int i = t; i < 1024; i += 256)
    ((float4*)sW)[i] = ((const float4*)W)[i];
  __syncthreads();

  int wave = t >> 5, lane = t & 31;
  int lane15 = lane & 15, khalf = lane >> 4;
  int tile = blockIdx.x * 8 + wave;
  if (tile * 16 >= nRows) return;           // wave-uniform: EXEC stays all-ones
  int m0 = tile * 16;

  v8f acc0 = {}, acc1 = {}, acc2 = {}, acc3 = {};
  const float* xrow = X + (size_t)(m0 + lane15) * 64;
  for (int k0 = 0; k0 < 64; k0 += 4) {
    int ka = k0 + 2 * khalf;
    v2f a; a.x = xrow[ka]; a.y = xrow[ka + 1];
    const float* wr0 = sW + ka * 64;
    const float* wr1 = wr0 + 64;
    v2f b0; b0.x = wr0[lane15 +  0]; b0.y = wr1[lane15 +  0];
    v2f b1; b1.x = wr0[lane15 + 16]; b1.y = wr1[lane15 + 16];
    v2f b2; b2.x = wr0[lane15 + 32]; b2.y = wr1[lane15 + 32];
    v2f b3; b3.x = wr0[lane15 + 48]; b3.y = wr1[lane15 + 48];
    acc0 = wmma_k4(a, b0, acc0);
    acc1 = wmma_k4(a, b1, acc1);
    acc2 = wmma_k4(a, b2, acc2);
    acc3 = wmma_k4(a, b3, acc3);
  }
  int rowBase = m0 + 8 * khalf;
  for (int r = 0; r < 8; ++r) {
    float* yr = Y + (size_t)(rowBase + r) * 64 + lane15;
    yr[ 0] = acc0[r];
    yr[16] = acc1[r];
    yr[32] = acc2[r];
    yr[48] = acc3[r];
  }
}

// --------------------------------------------- scatter: agg[dst] += xW[src]
// 16 threads per edge, float4 chunks, native f32 L2 atomics
__global__ __launch_bounds__(256) void scatter_kernel(const float* __restrict__ xW,
                                                      const int* __restrict__ src,
                                                      const int* __restrict__ dst,
                                                      float* __restrict__ agg, int nE) {
  int t = blockIdx.x * 256 + threadIdx.x;
  int e = t >> 4;
  if (e >= nE) return;
  int c = (t & 15) * 4;
  int s = src[e], d = dst[e];
  float4 v = *(const float4*)(xW + (size_t)s * 64 + c);
  float* a = agg + (size_t)d * 64 + c;
  atomicAddF32(a + 0, v.x);
  atomicAddF32(a + 1, v.y);
  atomicAddF32(a + 2, v.z);
  atomicAddF32(a + 3, v.w);
}

// --------- H = relu( agg/max(cnt,1) + X @ Root + b )   (Root GEMM via WMMA)
__global__ __launch_bounds__(256) void combine_kernel(const float* __restrict__ X,
                                                      const float* __restrict__ Root,
                                                      const float* __restrict__ bias,
                                                      const float* __restrict__ agg,
                                                      const float* __restrict__ cnt,
                                                      float* __restrict__ H, int nRows) {
  __shared__ float sW[64 * 64];
  __shared__ float sB[64];
  int t = threadIdx.x;
  for (int i = t; i < 1024; i += 256)
    ((float4*)sW)[i] = ((const float4*)Root)[i];
  if (t < 64) sB[t] = bias[t];
  __syncthreads();

  int wave = t >> 5, lane = t & 31;
  int lane15 = lane & 15, khalf = lane >> 4;
  int tile = blockIdx.x * 8 + wave;
  if (tile * 16 >= nRows) return;
  int m0 = tile * 16;

  v8f acc0 = {}, acc1 = {}, acc2 = {}, acc3 = {};
  const float* xrow = X + (size_t)(m0 + lane15) * 64;
  for (int k0 = 0; k0 < 64; k0 += 4) {
    int ka = k0 + 2 * khalf;
    v2f a; a.x = xrow[ka]; a.y = xrow[ka + 1];
    const float* wr0 = sW + ka * 64;
    const float* wr1 = wr0 + 64;
    v2f b0; b0.x = wr0[lane15 +  0]; b0.y = wr1[lane15 +  0];
    v2f b1; b1.x = wr0[lane15 + 16]; b1.y = wr1[lane15 + 16];
    v2f b2; b2.x = wr0[lane15 + 32]; b2.y = wr1[lane15 + 32];
    v2f b3; b3.x = wr0[lane15 + 48]; b3.y = wr1[lane15 + 48];
    acc0 = wmma_k4(a, b0, acc0);
    acc1 = wmma_k4(a, b1, acc1);
    acc2 = wmma_k4(a, b2, acc2);
    acc3 = wmma_k4(a, b3, acc3);
  }
  int rowBase = m0 + 8 * khalf;
  for (int r = 0; r < 8; ++r) {
    int row = rowBase + r;
    float inv = 1.0f / fmaxf(cnt[row], 1.0f);
    const float* ar = agg + (size_t)row * 64 + lane15;
    float* hr = H + (size_t)row * 64 + lane15;
    hr[ 0] = fmaxf(acc0[r] + ar[ 0] * inv + sB[lane15 +  0], 0.0f);
    hr[16] = fmaxf(acc1[r] + ar[16] * inv + sB[lane15 + 16], 0.0f);
    hr[32] = fmaxf(acc2[r] + ar[32] * inv + sB[lane15 + 32], 0.0f);
    hr[48] = fmaxf(acc3[r] + ar[48] * inv + sB[lane15 + 48], 0.0f);
  }
}

// ------------------------------------- graph mean-pool accumulation (atomics)
__global__ __launch_bounds__(256) void pool_kernel(const float* __restrict__ H,
                                                   const int* __restrict__ batch,
                                                   float* __restrict__ pooled,
                                                   float* __restrict__ gcnt, int nRows) {
  int t = blockIdx.x * 256 + threadIdx.x;
  int i = t >> 4;
  if (i >= nRows) return;
  int c = (t & 15) * 4;
  int g = batch[i];
  float4 v = *(const float4*)(H + (size_t)i * 64 + c);
  float* p = pooled + (size_t)g * 64 + c;
  atomicAddF32(p + 0, v.x);
  atomicAddF32(p + 1, v.y);
  atomicAddF32(p + 2, v.z);
  atomicAddF32(p + 3, v.w);
  if ((t & 15) == 0) atomicAddF32(&gcnt[g], 1.0f);
}

// ------------- final heads: mean, hidden/cell GEMMs, logits, log_softmax
// single block, 8 waves; hidden/cell each by 4 waves via WMMA from LDS
__global__ __launch_bounds__(256) void head_kernel(const float* __restrict__ pooledSum,
                                                   const float* __restrict__ gcnt,
                                                   const float* __restrict__ Wh,
                                                   const float* __restrict__ bh,
                                                   const float* __restrict__ Wc,
                                                   const float* __restrict__ bc,
                                                   const float* __restrict__ Wo,
                                                   const float* __restrict__ bo,
                                                   float* __restrict__ out) {
  __shared__ float sP[64 * 64];
  __shared__ float sHid[64 * 64];
  __shared__ float sLog[64 * 32];
  int t = threadIdx.x;
  for (int i = t; i < 4096; i += 256)
    sP[i] = pooledSum[i] / fmaxf(gcnt[i >> 6], 1.0f);
  __syncthreads();

  int wave = t >> 5, lane = t & 31;
  int lane15 = lane & 15, khalf = lane >> 4;

  // waves 0-3: hidden = P @ Wh + bh ; waves 4-7: cell = P @ Wc + bc
  {
    int rtile = wave & 3;
    bool isCell = wave >= 4;
    const float* W = isCell ? Wc : Wh;
    const float* b = isCell ? bc : bh;
    int m0 = rtile * 16;
    v8f acc0 = {}, acc1 = {}, acc2 = {}, acc3 = {};
    const float* arow = sP + (m0 + lane15) * 64;
    for (int k0 = 0; k0 < 64; k0 += 4) {
      int ka = k0 + 2 * khalf;
      v2f a; a.x = arow[ka]; a.y = arow[ka + 1];
      const float* wr0 = W + ka * 64;
      const float* wr1 = wr0 + 64;
      v2f b0; b0.x = wr0[lane15 +  0]; b0.y = wr1[lane15 +  0];
      v2f b1; b1.x = wr0[lane15 + 16]; b1.y = wr1[lane15 + 16];
      v2f b2; b2.x = wr0[lane15 + 32]; b2.y = wr1[lane15 + 32];
      v2f b3; b3.x = wr0[lane15 + 48]; b3.y = wr1[lane15 + 48];
      acc0 = wmma_k4(a, b0, acc0);
      acc1 = wmma_k4(a, b1, acc1);
      acc2 = wmma_k4(a, b2, acc2);
      acc3 = wmma_k4(a, b3, acc3);
    }
    float* dst = out + (isCell ? (2048 + 4096) : 2048);
    int rowBase = m0 + 8 * khalf;
    for (int r = 0; r < 8; ++r) {
      int row = rowBase + r;
      float v0 = acc0[r] + b[lane15 +  0];
      float v1 = acc1[r] + b[lane15 + 16];
      float v2 = acc2[r] + b[lane15 + 32];
      float v3 = acc3[r] + b[lane15 + 48];
      dst[row * 64 + lane15 +  0] = v0;
      dst[row * 64 + lane15 + 16] = v1;
      dst[row * 64 + lane15 + 32] = v2;
      dst[row * 64 + lane15 + 48] = v3;
      if (!isCell) {
        sHid[row * 64 + lane15 +  0] = v0;
        sHid[row * 64 + lane15 + 16] = v1;
        sHid[row * 64 + lane15 + 32] = v2;
        sHid[row * 64 + lane15 + 48] = v3;
      }
    }
  }
  __syncthreads();

  // waves 0-3: logits_pre = hidden @ Wo + bo   (64x64 @ 64x32)
  if (wave < 4) {
    int m0 = wave * 16;
    v8f acc0 = {}, acc1 = {};
    const float* arow = sHid + (m0 + lane15) * 64;
    for (int k0 = 0; k0 < 64; k0 += 4) {
      int ka = k0 + 2 * khalf;
      v2f a; a.x = arow[ka]; a.y = arow[ka + 1];
      const float* wr0 = Wo + ka * 32;
      const float* wr1 = wr0 + 32;
      v2f b0; b0.x = wr0[lane15 +  0]; b0.y = wr1[lane15 +  0];
      v2f b1; b1.x = wr0[lane15 + 16]; b1.y = wr1[lane15 + 16];
      acc0 = wmma_k4(a, b0, acc0);
      acc1 = wmma_k4(a, b1, acc1);
    }
    int rowBase = m0 + 8 * khalf;
    for (int r = 0; r < 8; ++r) {
      int row = rowBase + r;
      sLog[row * 32 + lane15 +  0] = acc0[r] + bo[lane15 +  0];
      sLog[row * 32 + lane15 + 16] = acc1[r] + bo[lane15 + 16];
    }
  }
  __syncthreads();

  // log_softmax per row; one wave32 covers exactly one row of 32 logits
  for (int r = 0; r < 8; ++r) {
    int row = wave * 8 + r;
    float v = sLog[row * 32 + lane];
    float m = v;
    for (int off = 16; off; off >>= 1) m = fmaxf(m, __shfl_xor(m, off));
    float ev = __expf(v - m);
    float s = ev;
    for (int off = 16; off; off >>= 1) s += __shfl_xor(s, off);
    out[row * 32 + lane] = v - m - __logf(s);
  }
}

extern "C" void kernel_launch(void* const* d_in, const int* in_sizes, int n_in,
                              void* d_out, int out_size, void* d_ws, size_t ws_size,
                              hipStream_t stream) {
  const float* x     = (const float*)d_in[0];   // [N,64]
  const int*   ei    = (const int*)d_in[1];     // [2,E] (int32 under default jax x64-off)
  const int*   batch = (const int*)d_in[2];     // [N]
  const float* W1    = (const float*)d_in[3];
  const float* root1 = (const float*)d_in[4];
  const float* b1    = (const float*)d_in[5];
  const float* W2    = (const float*)d_in[6];
  const float* root2 = (const float*)d_in[7];
  const float* b2    = (const float*)d_in[8];
  const float* Wh    = (const float*)d_in[9];
  const float* bh    = (const float*)d_in[10];
  const float* Wc    = (const float*)d_in[11];
  const float* bc    = (const float*)d_in[12];
  const float* Wo    = (const float*)d_in[13];
  const float* bo    = (const float*)d_in[14];
  float* out = (float*)d_out;

  const int N = in_sizes[0] / 64;       // 100000 (multiple of 16)
  const int E = in_sizes[1] / 2;        // 1600000
  const int G = G_GRAPHS;

  // workspace layout (floats)
  float* bufA   = (float*)d_ws;             // [N,64]  xW / h2
  float* bufB   = bufA + (size_t)N * 64;    // [N,64]  agg (atomics target)
  float* bufC   = bufB + (size_t)N * 64;    // [N,64]  h1
  float* cnt    = bufC + (size_t)N * 64;    // [N]
  float* pooled = cnt + N;                  // [G,64]
  float* gcnt   = pooled + (size_t)G * 64;  // [G]

  const int nodeVec  = N * 64;
  const int smallLen = N + G * 64 + G;      // cnt + pooled + gcnt (contiguous)
  const int gemmBlocks    = (N / 16 + 7) / 8;
  const int scatterBlocks = (E * 16 + 255) / 256;
  const int poolBlocks    = (N * 16 + 255) / 256;

  // init accumulators
  zero_kernel<<<(nodeVec + 255) / 256, 256, 0, stream>>>(bufB, nodeVec);
  zero_kernel<<<(smallLen + 255) / 256, 256, 0, stream>>>(cnt, smallLen);

  // in-degree (shared by both layers)
  degree_kernel<<<(E + 255) / 256, 256, 0, stream>>>(ei + E, cnt, E);

  // ---- layer 1
  gemm64_kernel<<<gemmBlocks, 256, 0, stream>>>(x, W1, bufA, N);
  scatter_kernel<<<scatterBlocks, 256, 0, stream>>>(bufA, ei, ei + E, bufB, E);
  combine_kernel<<<gemmBlocks, 256, 0, stream>>>(x, root1, b1, bufB, cnt, bufC, N);

  // ---- layer 2
  zero_kernel<<<(nodeVec + 255) / 256, 256, 0, stream>>>(bufB, nodeVec);
  gemm64_kernel<<<gemmBlocks, 256, 0, stream>>>(bufC, W2, bufA, N);
  scatter_kernel<<<scatterBlocks, 256, 0, stream>>>(bufA, ei, ei + E, bufB, E);
  combine_kernel<<<gemmBlocks, 256, 0, stream>>>(bufC, root2, b2, bufB, cnt, bufA, N);

  // ---- pooling + heads
  pool_kernel<<<poolBlocks, 256, 0, stream>>>(bufA, batch, pooled, gcnt, N);
  head_kernel<<<1, 256, 0, stream>>>(pooled, gcnt, Wh, bh, Wc, bc, Wo, bo, out);
}